// SentenceMatrixLayer_1580547969803
// MI455X (gfx1250) — compile-verified
//
#include <hip/hip_runtime.h>

typedef __attribute__((ext_vector_type(2))) float v2f;
typedef __attribute__((ext_vector_type(4))) float f4;
typedef __attribute__((ext_vector_type(8))) float v8f;

#define D_IN   128
#define N_TOK  2048
#define P_ASEM 0.8f

// ---------------------------------------------------------------------------
// Kernel 1: dual-GEMV via V_WMMA_F32_16X16X4_F32.
// Each wave32 handles 16 rows of x (D=128). A-tile = 16x4 chunk of x,
// B-tile = 4x16 with col0 = W[:D] chunk, col1 = W[D:] chunk, cols 2..15 = 0.
// After accumulating over K=128, D-matrix col0 = si, col1 = sj.
//
// f32 A 16x4 layout : lanes 0-15 -> K={0,1} in VGPR{0,1}; lanes 16-31 -> K={2,3}
// f32 B 4x16 layout : lanes 0-15 (N=lane) -> K={0,1}; lanes 16-31 -> K={2,3}
// f32 C 16x16 layout: VGPR v, lanes 0-15 -> (M=v, N=lane); lanes 16-31 -> (M=v+8)
// ---------------------------------------------------------------------------
__global__ __launch_bounds__(256) void rowscore_wmma(const float* __restrict__ x,
                                                     const float* __restrict__ W,
                                                     float* __restrict__ si,
                                                     float* __restrict__ sj)
{
    const int lane    = threadIdx.x & 31;
    const int wave    = blockIdx.x * (blockDim.x >> 5) + (threadIdx.x >> 5);
    const int rowBase = wave << 4;                 // 16 rows per wave
    const int m       = lane & 15;
    const int kh      = lane >> 4;                 // which K half this lane holds
    const int n       = lane & 15;                 // B/N column this lane holds

    const float* xrow = x + (size_t)(rowBase + m) * D_IN;
    const float* Wr   = W;                         // W[:D]   -> column 0 (si)
    const float* Wc   = W + D_IN;                  // W[D:2D] -> column 1 (sj)

    v8f c = {0.f, 0.f, 0.f, 0.f, 0.f, 0.f, 0.f, 0.f};

#pragma unroll 4
    for (int kc = 0; kc < D_IN / 4; ++kc) {
        const int k0 = kc * 4 + kh * 2;            // this lane's K pair
        v2f a  = *(const v2f*)(xrow + k0);
        v2f wr = *(const v2f*)(Wr + k0);
        v2f wc = *(const v2f*)(Wc + k0);
        v2f bm;
        bm.x = (n == 0) ? wr.x : ((n == 1) ? wc.x : 0.0f);
        bm.y = (n == 0) ? wr.y : ((n == 1) ? wc.y : 0.0f);
        // (neg_a, A, neg_b, B, c_mod, C, reuse_a, reuse_b)
        c = __builtin_amdgcn_wmma_f32_16x16x4_f32(false, a, false, bm,
                                                  (short)0, c, false, false);
    }

    // Column 0 (si): lane 0 holds rows 0..7, lane 16 holds rows 8..15.
    // Column 1 (sj): lane 1 holds rows 0..7, lane 17 holds rows 8..15.
    float* dst = nullptr;
    int    off = 0;
    if      (lane == 0)  { dst = si; off = rowBase;     }
    else if (lane == 16) { dst = si; off = rowBase + 8; }
    else if (lane == 1)  { dst = sj; off = rowBase;     }
    else if (lane == 17) { dst = sj; off = rowBase + 8; }
    if (dst) {
#pragma unroll
        for (int v = 0; v < 8; ++v) dst[off + v] = c[v];
    }
}

// ---------------------------------------------------------------------------
// Kernel 2: bandwidth-bound blend. out[b,i,j] = 0.8*sigmoid(si+sj+b) + 0.2*adj
// One float4 per lane; nontemporal b128 for the 134MB+134MB adj/out streams
// (they exceed the 192MB L2, keep si/sj hot instead).
// ---------------------------------------------------------------------------
__global__ __launch_bounds__(256) void blend_kernel(const float* __restrict__ adj,
                                                    const float* __restrict__ si,
                                                    const float* __restrict__ sj,
                                                    const float* __restrict__ bptr,
                                                    float* __restrict__ out,
                                                    int total4)
{
    const int g = blockIdx.x * blockDim.x + threadIdx.x;
    if (g >= total4) return;

    const float  bias = *bptr;
    const size_t e    = (size_t)g << 2;            // flat element index
    const int    row  = (int)(e >> 11);            // b*N + i   (N = 2048 = 2^11)
    const int    bidx = (int)(e >> 22);            // batch     (N*N = 2^22)
    const int    j    = (int)(e & (size_t)(N_TOK - 1));

    const float siv  = si[row];
    const f4    sj4  = *(const f4*)(sj + ((size_t)bidx << 11) + j);
    const f4    adj4 = __builtin_nontemporal_load((const f4*)(adj + e));

    f4 o;
#pragma unroll
    for (int k = 0; k < 4; ++k) {
        const float s   = siv + sj4[k] + bias;
        const float sig = 1.0f / (1.0f + __expf(-s));
        o[k] = P_ASEM * sig + (1.0f - P_ASEM) * adj4[k];
    }
    __builtin_nontemporal_store(o, (f4*)(out + e));
}

// ---------------------------------------------------------------------------
extern "C" void kernel_launch(void* const* d_in, const int* in_sizes, int n_in,
                              void* d_out, int out_size, void* d_ws, size_t ws_size,
                              hipStream_t stream)
{
    const float* x   = (const float*)d_in[0];   // [B, N, D]
    const float* adj = (const float*)d_in[1];   // [B, N, N]
    const float* W   = (const float*)d_in[2];   // [2D]
    const float* b   = (const float*)d_in[3];   // scalar
    float*       out = (float*)d_out;           // [B, N, N]

    const int BN = in_sizes[0] / D_IN;          // B*N rows (16384)
    float* si = (float*)d_ws;                   // BN floats
    float* sj = si + BN;                        // BN floats (128 KB total)

    // Phase A: 16 rows per wave, 8 waves (256 threads) per block.
    const int waves   = BN / 16;                // 1024
    const int blocks1 = waves / 8;              // 128
    rowscore_wmma<<<blocks1, 256, 0, stream>>>(x, W, si, sj);

    // Phase B: streaming blend, one float4 per lane.
    const int total4  = out_size / 4;           // 8,388,608
    const int blocks2 = (total4 + 255) / 256;   // 32,768
    blend_kernel<<<blocks2, 256, 0, stream>>>(adj, si, sj, b, out, total4);
}